// SkeletonRecallLoss_44719199486537
// MI455X (gfx1250) — compile-verified
//
#include <hip/hip_runtime.h>
#include <stdint.h>

// ---------------------------------------------------------------------------
// Skeleton Recall Loss, MI455X (gfx1250, wave32)
//  - bit-packed binary erosion (AND of 27-neighborhood): 9 passes over 1MB,
//    fully L2-resident (192MB L2) instead of ~2GB of f32 min-pool traffic
//  - c(v) = sum of 10 nested masks; loss = 1 - sum(c*p)/(sum(c)+cmax*1e-5)
//  - v_wmma_f32_16x16x4_f32 for exact f32 wave-level sum reduction
//  - TDM (tensor_load_to_lds + s_wait_tensorcnt) double-buffers logits
//    chunks into LDS, overlapping DMA with compute
// ---------------------------------------------------------------------------

static constexpr int kDim     = 160;
static constexpr int kNB      = kDim * kDim * kDim;   // 4,096,000 voxels/batch
static constexpr int kNV      = 2 * kNB;              // 8,192,000 voxels total
static constexpr int kWords   = kNV / 32;             // 256,000 packed words
static constexpr int kPlanes  = 10;                   // gt + 9 erosions
static constexpr int kChunk   = 2048;                 // voxels per TDM chunk
static constexpr int kNChunks = kNV / kChunk;         // 4000 (2000/batch, exact)
static constexpr int kDotBlks = 1024;

typedef __attribute__((ext_vector_type(2))) float v2f;
typedef __attribute__((ext_vector_type(8))) float v8f;
typedef __attribute__((ext_vector_type(4))) unsigned tdm_v4u;
typedef __attribute__((ext_vector_type(8))) int      tdm_v8i;
typedef __attribute__((ext_vector_type(4))) int      tdm_v4i;

// ---- wave32 sum reduction via V_WMMA_F32_16X16X4_F32 ----------------------
// A: 16x4 f32 (a.x = lane value, a.y = 0), B: 4x16 all-ones.
// D[m][n] = x(lane m) + x(lane m+16); lanes 0-15 hold rows 0-7 in c[0..7],
// lanes 16-31 rows 8-15, so (own 8-sum) + (xor-16 partner) = full wave sum.
__device__ __forceinline__ float waveReduceSum(float x) {
#if __has_builtin(__builtin_amdgcn_wmma_f32_16x16x4_f32)
  v2f a; a.x = x;    a.y = 0.0f;
  v2f b; b.x = 1.0f; b.y = 1.0f;
  v8f c = {};
  c = __builtin_amdgcn_wmma_f32_16x16x4_f32(false, a, false, b, (short)0, c,
                                            false, false);
  float s = c[0] + c[1] + c[2] + c[3] + c[4] + c[5] + c[6] + c[7];
  return s + __shfl_xor(s, 16, 32);
#else
  for (int off = 16; off; off >>= 1) x += __shfl_xor(x, off, 32);
  return x;
#endif
}

__device__ __forceinline__ int waveReduceMax(int x) {
  for (int off = 16; off; off >>= 1) {
    int o = __shfl_xor(x, off, 32);
    x = o > x ? o : x;
  }
  return x;
}

// ---- kernel 1: pack gt bits (one wave32 ballot per 32 voxels) -------------
__global__ void prep_mask_kernel(const int* __restrict__ labels,
                                 uint32_t* __restrict__ plane0) {
  const int v = blockIdx.x * 256 + threadIdx.x;   // grid exact (kNV/256)
  const int lab = labels[v];
#if __has_builtin(__builtin_amdgcn_ballot_w32)
  const unsigned m = __builtin_amdgcn_ballot_w32(lab > 0);
#else
  const unsigned m = (unsigned)__ballot(lab > 0);
#endif
  if ((threadIdx.x & 31) == 0) plane0[v >> 5] = m;
}

// ---- kernel 2: one bit-packed 3x3x3 binary erosion pass -------------------
// word idx = ((b*160+z)*160+y)*5 + w ; pad value +inf == bit 1 for AND.
__global__ void erode_kernel(const uint32_t* __restrict__ src,
                             uint32_t* __restrict__ dst) {
  const int idx = blockIdx.x * 256 + threadIdx.x; // grid exact (kWords/256)
  const int w   = idx % 5;
  const int row = idx / 5;        // (b*160+z)*160 + y
  const int y   = row % kDim;
  const int zz  = row / kDim;     // b*160 + z
  const int z   = zz % kDim;
  uint32_t acc = 0xFFFFFFFFu;
#pragma unroll
  for (int dz = -1; dz <= 1; ++dz) {
    if ((unsigned)(z + dz) >= (unsigned)kDim) continue;  // OOB row == all 1s
#pragma unroll
    for (int dy = -1; dy <= 1; ++dy) {
      if ((unsigned)(y + dy) >= (unsigned)kDim) continue;
      const uint32_t* r = src + ((size_t)(zz + dz) * kDim + (y + dy)) * 5;
      const uint32_t v  = r[w];
      const uint32_t lo = (w == 0) ? 1u          : (r[w - 1] >> 31); // x-1 pad
      const uint32_t hi = (w == 4) ? 0x80000000u : (r[w + 1] << 31); // x+1 pad
      acc &= v & ((v << 1) | lo) & ((v >> 1) | hi);
    }
  }
  dst[idx] = acc;
}

// ---- TDM: 1-D contiguous tile of `nelem` f32 -> LDS at lds_byte_off -------
__device__ __forceinline__ void tdm_load_chunk(const float* gptr,
                                               unsigned lds_byte_off,
                                               int nelem) {
  const uint64_t ga = (uint64_t)(uintptr_t)gptr;
  tdm_v4u g0;
  g0[0] = 1u;                                       // count=1 (user descriptor)
  g0[1] = lds_byte_off;                             // lds_addr (bytes)
  g0[2] = (unsigned)ga;                             // global_addr[31:0]
  g0[3] = ((unsigned)(ga >> 32) & 0x01FFFFFFu) | (2u << 30); // [56:32] | type=2
  tdm_v8i g1;
  g1[0] = (int)(2u << 16);                          // data_size=4B, no flags
  g1[1] = (int)(((unsigned)nelem & 0xFFFFu) << 16); // tensor_dim0[15:0]
  g1[2] = (int)(((unsigned)nelem >> 16) & 0xFFFFu); // tensor_dim0[31:16]
  g1[3] = (int)(((unsigned)nelem & 0xFFFFu) << 16); // tile_dim0 (1-D tile)
  g1[4] = 0;                                        // tile_dim1/2 unused
  g1[5] = (int)(unsigned)nelem;                     // dim0 stride (unused, 1-D)
  g1[6] = 0;
  g1[7] = 0;
  const tdm_v4i gz = {0, 0, 0, 0};
#if __clang_major__ >= 23
  const tdm_v8i gz8 = {0, 0, 0, 0, 0, 0, 0, 0};
  __builtin_amdgcn_tensor_load_to_lds(g0, g1, gz, gz, gz8, 0);
#else
  __builtin_amdgcn_tensor_load_to_lds(g0, g1, gz, gz, 0);
#endif
}

// issue both logit-channel loads of `chunk` into LDS buffer `buf`
__device__ __forceinline__ void issue_chunk(const float* __restrict__ logits,
                                            float (*lds)[2 * kChunk],
                                            int chunk, int buf) {
  const int    v0 = chunk * kChunk;
  const int    b  = (v0 >= kNB) ? 1 : 0;
  const float* l0 = logits + (size_t)v0 + (size_t)b * kNB;  // [b,0,...]
  const float* l1 = l0 + kNB;                               // [b,1,...]
  const unsigned base = (unsigned)(uintptr_t)(void*)&lds[buf][0]; // LDS offset
  tdm_load_chunk(l0, base, kChunk);
  tdm_load_chunk(l1, base + (unsigned)(kChunk * sizeof(float)), kChunk);
}

// ---- kernel 3: c(v) from 10 bitplanes, dot with sigmoid(l1-l0) ------------
__global__ void dot_kernel(const float* __restrict__ logits,
                           const uint32_t* __restrict__ planes,
                           float* __restrict__ pnum,
                           float* __restrict__ pden,
                           int* __restrict__ pmax) {
  __shared__ float lds[2][2 * kChunk];  // double-buffered {l0 | l1} chunks
  __shared__ float rn[8], rd[8];
  __shared__ int   rm[8];
  const int t    = threadIdx.x;
  const int lane = t & 31;
  float num = 0.f, den = 0.f;
  int   cm  = 0;

  // prologue: wave 0 launches the DMA for this block's first chunk
  if (t == 0) issue_chunk(logits, lds, blockIdx.x, 0);

  int parity = 0;
  for (int chunk = blockIdx.x; chunk < kNChunks; chunk += kDotBlks) {
    const int next = chunk + kDotBlks;
    if (t == 0) {
      if (next < kNChunks) {
        issue_chunk(logits, lds, next, parity ^ 1);  // prefetch into other buf
        __builtin_amdgcn_s_wait_tensorcnt(2);        // in-order: current done
      } else {
        __builtin_amdgcn_s_wait_tensorcnt(0);
      }
    }
    __syncthreads();  // chunk data visible to all waves

    const int v0 = chunk * kChunk;
#pragma unroll 4
    for (int k = 0; k < kChunk / 256; ++k) {
      const int li = t + k * 256;
      const int v  = v0 + li;
      const int wi = v >> 5;     // whole wave reads the same word (broadcast)
      int c = 0;
#pragma unroll
      for (int j = 0; j < kPlanes; ++j)
        c += (int)((planes[(size_t)j * kWords + wi] >> lane) & 1u);
      const float p =
          1.0f / (1.0f + __expf(lds[parity][li] - lds[parity][kChunk + li]));
      num += (float)c * p;
      den += (float)c;
      cm   = c > cm ? c : cm;
    }
    __syncthreads();  // all reads done before this buffer is DMA-overwritten
    parity ^= 1;
  }

  const float wn = waveReduceSum(num);   // v_wmma_f32_16x16x4_f32
  const float wd = waveReduceSum(den);
  const int   wm = waveReduceMax(cm);
  const int wave = t >> 5;
  if (lane == 0) { rn[wave] = wn; rd[wave] = wd; rm[wave] = wm; }
  __syncthreads();
  if (t == 0) {
    float sn = 0.f, sd = 0.f; int sm = 0;
    for (int i = 0; i < 8; ++i) { sn += rn[i]; sd += rd[i]; sm = rm[i] > sm ? rm[i] : sm; }
    pnum[blockIdx.x] = sn; pden[blockIdx.x] = sd; pmax[blockIdx.x] = sm;
  }
}

// ---- kernel 4: fixed-order final reduction -> scalar loss -----------------
__global__ void final_kernel(const float* __restrict__ pnum,
                             const float* __restrict__ pden,
                             const int* __restrict__ pmax,
                             float* __restrict__ out) {
  __shared__ float rn[8], rd[8];
  __shared__ int   rm[8];
  const int t = threadIdx.x;
  float num = 0.f, den = 0.f; int cm = 0;
  for (int i = t; i < kDotBlks; i += 256) {
    num += pnum[i]; den += pden[i];
    cm = pmax[i] > cm ? pmax[i] : cm;
  }
  const float wn = waveReduceSum(num);
  const float wd = waveReduceSum(den);
  const int   wm = waveReduceMax(cm);
  if ((t & 31) == 0) { rn[t >> 5] = wn; rd[t >> 5] = wd; rm[t >> 5] = wm; }
  __syncthreads();
  if (t == 0) {
    float sn = 0.f, sd = 0.f; int sm = 0;
    for (int i = 0; i < 8; ++i) { sn += rn[i]; sd += rd[i]; sm = rm[i] > sm ? rm[i] : sm; }
    // skel = c/10, skel_max = cmax/10 -> ratio = sum(c*p)/(sum(c)+cmax*1e-5)
    float loss = 1.0f;                       // skel_max==0 path of reference
    if (sd > 0.f) loss = 1.0f - sn / (sd + (float)sm * 1e-5f);
    out[0] = loss;
  }
}

// ---------------------------------------------------------------------------
extern "C" void kernel_launch(void* const* d_in, const int* in_sizes, int n_in,
                              void* d_out, int out_size, void* d_ws, size_t ws_size,
                              hipStream_t stream) {
  const float* logits = (const float*)d_in[0];  // (2,2,160,160,160) f32
  const int*   labels = (const int*)d_in[1];    // (2,1,160,160,160) i32

  uint32_t* planes = (uint32_t*)d_ws;           // 10 bitplanes: 10.24 MB
  char*  tail = (char*)d_ws + (size_t)kPlanes * kWords * sizeof(uint32_t);
  float* pnum = (float*)tail;
  float* pden = pnum + kDotBlks;
  int*   pmax = (int*)(pden + kDotBlks);

  prep_mask_kernel<<<kNV / 256, 256, 0, stream>>>(labels, planes);
  for (int j = 1; j < kPlanes; ++j)
    erode_kernel<<<kWords / 256, 256, 0, stream>>>(planes + (size_t)(j - 1) * kWords,
                                                   planes + (size_t)j * kWords);
  dot_kernel<<<kDotBlks, 256, 0, stream>>>(logits, planes, pnum, pden, pmax);
  final_kernel<<<1, 256, 0, stream>>>(pnum, pden, pmax, (float*)d_out);
}